// OrthogonalSylvesterVAE_84705345012413
// MI455X (gfx1250) — compile-verified
//
#include <hip/hip_runtime.h>
#include <hip/hip_bf16.h>
#include <stdint.h>

// ---------------------------------------------------------------------------
// OrthogonalSylvesterVAE for MI455X (gfx1250, wave32, WMMA f32 16x16x4,
// async global<->LDS data movers for tile staging)
// ---------------------------------------------------------------------------

#define BATCH 256
#define LATD  64
#define QOUTD 1024
#define KF    4
#define DDIM  32
#define ZDIM  64
#define NMAT  (BATCH * KF)      // 1024 matrices of 64x32
#define STEPS_ORTHO 100
#define BN_EPS 1e-5f

typedef float v2f __attribute__((ext_vector_type(2)));
typedef float v8f __attribute__((ext_vector_type(8)));

__device__ __forceinline__ v8f wmma4(v2f a, v2f b, v8f c) {
  // D(16x16,f32) = A(16x4,f32) x B(4x16,f32) + C
  return __builtin_amdgcn_wmma_f32_16x16x4_f32(false, a, false, b, (short)0, c,
                                               false, false);
}

// --- async global<->LDS movers (ASYNCcnt-tracked), per CDNA5 ISA --------
__device__ __forceinline__ void async_g2l_b32(const float* gptr, void* lptr) {
  uint32_t lds_off = (uint32_t)(uintptr_t)lptr;   // flat LDS addr[31:0] = LDS offset
  uint64_t gaddr   = (uint64_t)(uintptr_t)gptr;
  asm volatile("global_load_async_to_lds_b32 %0, %1, off"
               :: "v"(lds_off), "v"(gaddr) : "memory");
}
__device__ __forceinline__ void async_l2g_b32(float* gptr, const void* lptr) {
  uint32_t lds_off = (uint32_t)(uintptr_t)lptr;
  uint64_t gaddr   = (uint64_t)(uintptr_t)gptr;
  asm volatile("global_store_async_from_lds_b32 %0, %1, off"
               :: "v"(gaddr), "v"(lds_off) : "memory");
}
__device__ __forceinline__ void async_wait0() {
  asm volatile("s_wait_asynccnt 0" ::: "memory");
}

// ---------------------------------------------------------------------------
// Generic f32 WMMA GEMM:  C[M,N] = A[M,K] @ B[K,N] + bias[N]   (mode1: tanh)
// One wave (32 threads) computes one 16x16 tile of C.
// ---------------------------------------------------------------------------
__global__ __launch_bounds__(32) void gemm_wmma_f32(
    const float* __restrict__ A, const float* __restrict__ B,
    const float* __restrict__ bias, float* __restrict__ C,
    int M, int N, int K, int mode) {
  int ntn  = N >> 4;
  int tile = blockIdx.x;
  int m0   = (tile / ntn) << 4;
  int n0   = (tile % ntn) << 4;
  int l    = threadIdx.x;
  int half = l >> 4;        // 0 for lanes 0-15, 1 for lanes 16-31
  int lm   = l & 15;

  v8f acc = {0.f, 0.f, 0.f, 0.f, 0.f, 0.f, 0.f, 0.f};
  const float* arow = A + (size_t)(m0 + lm) * K;
  for (int k = 0; k < K; k += 4) {
    int kk = k + 2 * half;
    v2f af, bf;
    af.x = arow[kk];
    af.y = arow[kk + 1];
    bf.x = B[(size_t)kk * N + n0 + lm];
    bf.y = B[(size_t)(kk + 1) * N + n0 + lm];
    acc = wmma4(af, bf, acc);
  }
  float bv = bias[n0 + lm];
#pragma unroll
  for (int r = 0; r < 8; r++) {
    float v = acc[r] + bv;
    if (mode == 1) v = tanhf(v);
    C[(size_t)(m0 + r + 8 * half) * N + n0 + lm] = v;
  }
}

// ---------------------------------------------------------------------------
// Orthogonalization scan: 100 steps of  A <- A @ (1.5 I - 0.5 A^T A)
// per 64x32 matrix.  One wave per matrix, 2 matrices per block, A in LDS.
// Tile staging uses the async global<->LDS data movers.
// ---------------------------------------------------------------------------
__global__ __launch_bounds__(64) void ortho_scan(float* __restrict__ amat) {
  __shared__ float As[2][64][33];   // 33: bank-conflict padding
  __shared__ float Ms[2][32][33];
  int wv   = threadIdx.x >> 5;
  int l    = threadIdx.x & 31;
  int half = l >> 4;
  int lm   = l & 15;
  int n    = blockIdx.x * 2 + wv;

  float (*A)[33]  = As[wv];
  float (*Mm)[33] = Ms[wv];
  float* g = amat + (size_t)n * (ZDIM * DDIM);

  // async preload HBM -> LDS (with padding remap), ASYNCcnt-tracked
  for (int i = l; i < ZDIM * DDIM; i += 32)
    async_g2l_b32(g + i, &A[i >> 5][i & 31]);
  async_wait0();
  __syncthreads();

#pragma unroll 1
  for (int s = 0; s < STEPS_ORTHO; s++) {
    // ---- gram = A^T A  (32x32 as 2x2 tiles of 16x16), K = 64 ----
    v8f g00 = {0.f,0.f,0.f,0.f,0.f,0.f,0.f,0.f};
    v8f g01 = g00, g10 = g00, g11 = g00;
#pragma unroll 1
    for (int zb = 0; zb < ZDIM; zb += 4) {
      int zr = zb + 2 * half;
      v2f f0, f1;                       // shared A/B fragments of A^T A
      f0.x = A[zr][lm];        f0.y = A[zr + 1][lm];
      f1.x = A[zr][16 + lm];   f1.y = A[zr + 1][16 + lm];
      g00 = wmma4(f0, f0, g00);
      g01 = wmma4(f0, f1, g01);
      g10 = wmma4(f1, f0, g10);
      g11 = wmma4(f1, f1, g11);
    }
    // ---- M = 1.5 I - 0.5 gram  (write from D-fragment layout) ----
#pragma unroll
    for (int r = 0; r < 8; r++) {
      int mr0 = r + 8 * half;
      int mc0 = lm, mc1 = 16 + lm;
      Mm[mr0][mc0]      = (mr0 == mc0 ? 1.5f : 0.f) - 0.5f * g00[r];
      Mm[mr0][mc1]      =                     0.f   - 0.5f * g01[r];
      Mm[16 + mr0][mc0] =                     0.f   - 0.5f * g10[r];
      Mm[16 + mr0][mc1] = (mr0 == lm  ? 1.5f : 0.f) - 0.5f * g11[r];
    }
    __syncthreads();

    // ---- A_new = A @ M  (64x32 as 4x2 tiles), K = 32 ----
    v8f u[4][2];
#pragma unroll
    for (int mt = 0; mt < 4; mt++)
#pragma unroll
      for (int nt = 0; nt < 2; nt++)
        u[mt][nt] = (v8f){0.f,0.f,0.f,0.f,0.f,0.f,0.f,0.f};
#pragma unroll 1
    for (int kb = 0; kb < DDIM; kb += 4) {
      int kr = kb + 2 * half;
      v2f am[4], bm[2];
#pragma unroll
      for (int mt = 0; mt < 4; mt++) {
        am[mt].x = A[mt * 16 + lm][kr];
        am[mt].y = A[mt * 16 + lm][kr + 1];
      }
#pragma unroll
      for (int nt = 0; nt < 2; nt++) {
        bm[nt].x = Mm[kr][nt * 16 + lm];
        bm[nt].y = Mm[kr + 1][nt * 16 + lm];
      }
#pragma unroll
      for (int mt = 0; mt < 4; mt++)
#pragma unroll
        for (int nt = 0; nt < 2; nt++)
          u[mt][nt] = wmma4(am[mt], bm[nt], u[mt][nt]);
    }
    __syncthreads();   // all reads of A done before overwrite
#pragma unroll
    for (int mt = 0; mt < 4; mt++)
#pragma unroll
      for (int nt = 0; nt < 2; nt++)
#pragma unroll
        for (int r = 0; r < 8; r++)
          A[mt * 16 + r + 8 * half][nt * 16 + lm] = u[mt][nt][r];
    __syncthreads();
  }

  // async store LDS -> HBM
  for (int i = l; i < ZDIM * DDIM; i += 32)
    async_l2g_b32(g + i, &A[i >> 5][i & 31]);
  async_wait0();
}

// ---------------------------------------------------------------------------
// Small VALU kernels
// ---------------------------------------------------------------------------
__global__ void conv2d_k(const float* __restrict__ in, const float* __restrict__ w,
                         const float* __restrict__ bias, float* __restrict__ out,
                         int Bn, int Cin, int Hin, int Win, int Cout, int KH,
                         int KW, int Hout) {
  int idx = blockIdx.x * blockDim.x + threadIdx.x;
  int total = Bn * Cout * Hout;
  if (idx >= total) return;
  int h  = idx % Hout;
  int co = (idx / Hout) % Cout;
  int b  = idx / (Hout * Cout);
  float acc = bias[co];
  for (int ci = 0; ci < Cin; ci++)
    for (int kh = 0; kh < KH; kh++)
      for (int kw = 0; kw < KW; kw++)
        acc += in[((b * Cin + ci) * Hin + h + kh) * Win + kw] *
               w[((co * Cin + ci) * KH + kh) * KW + kw];
  out[idx] = acc;
}

__global__ void tconv2d_k(const float* __restrict__ in, const float* __restrict__ wgt,
                          const float* __restrict__ bias, float* __restrict__ out,
                          int Bn, int Cin, int Hin, int Win, int Cout, int KH,
                          int KW, int Hout, int Wout) {
  int idx = blockIdx.x * blockDim.x + threadIdx.x;
  int total = Bn * Cout * Hout * Wout;
  if (idx >= total) return;
  int wo = idx % Wout;
  int h  = (idx / Wout) % Hout;
  int ci = (idx / (Wout * Hout)) % Cout;
  int b  = idx / (Wout * Hout * Cout);
  float acc = bias[ci];
  for (int o = 0; o < Cin; o++)
    for (int kh = 0; kh < KH; kh++) {
      int hi = h - kh;
      if (hi < 0 || hi >= Hin) continue;
      for (int kw = 0; kw < KW; kw++) {
        int wi = wo - kw;
        if (wi < 0 || wi >= Win) continue;
        acc += in[((b * Cin + o) * Hin + hi) * Win + wi] *
               wgt[((o * Cout + ci) * KH + kh) * KW + kw];
      }
    }
  out[idx] = acc;
}

__global__ __launch_bounds__(256) void bn2d_relu_k(
    const float* __restrict__ x, const float* __restrict__ g,
    const float* __restrict__ be, float* __restrict__ y, int Bn, int C, int HW) {
  int c = blockIdx.x, t = threadIdx.x;
  int N = Bn * HW;
  __shared__ float rs[256], rs2[256];
  float s = 0.f, s2 = 0.f;
  for (int i = t; i < N; i += 256) {
    int b = i / HW, hw = i - b * HW;
    float v = x[(b * C + c) * HW + hw];
    s += v; s2 += v * v;
  }
  rs[t] = s; rs2[t] = s2;
  __syncthreads();
  for (int o = 128; o > 0; o >>= 1) {
    if (t < o) { rs[t] += rs[t + o]; rs2[t] += rs2[t + o]; }
    __syncthreads();
  }
  float mean = rs[0] / N;
  float var  = rs2[0] / N - mean * mean;
  float sc   = g[c] * rsqrtf(var + BN_EPS);
  float sh   = be[c] - mean * sc;
  for (int i = t; i < N; i += 256) {
    int b = i / HW, hw = i - b * HW;
    int ix = (b * C + c) * HW + hw;
    y[ix] = fmaxf(0.f, x[ix] * sc + sh);
  }
}

__global__ void bn1d_relu_k(const float* __restrict__ x, const float* __restrict__ g,
                            const float* __restrict__ be, float* __restrict__ y,
                            int Bn, int F) {
  int f = blockIdx.x * blockDim.x + threadIdx.x;
  if (f >= F) return;
  float s = 0.f, s2 = 0.f;
  for (int b = 0; b < Bn; b++) { float v = x[b * F + f]; s += v; s2 += v * v; }
  float mean = s / Bn;
  float var  = s2 / Bn - mean * mean;
  float sc   = g[f] * rsqrtf(var + BN_EPS);
  float sh   = be[f] - mean * sc;
  for (int b = 0; b < Bn; b++) {
    int ix = b * F + f;
    y[ix] = fmaxf(0.f, x[ix] * sc + sh);
  }
}

__global__ __launch_bounds__(256) void row_norm_k(float* __restrict__ q) {
  int r = blockIdx.x, t = threadIdx.x;
  float* row = q + (size_t)r * (ZDIM * DDIM);
  __shared__ float red[256];
  float s = 0.f;
  for (int i = t; i < ZDIM * DDIM; i += 256) { float v = row[i]; s += v * v; }
  red[t] = s;
  __syncthreads();
  for (int o = 128; o > 0; o >>= 1) {
    if (t < o) red[t] += red[t + o];
    __syncthreads();
  }
  float inv = rsqrtf(red[0]);
  for (int i = t; i < ZDIM * DDIM; i += 256) row[i] *= inv;
}

__global__ void z0_init_k(const float* __restrict__ mean,
                          const float* __restrict__ logvar,
                          const float* __restrict__ eps, float* __restrict__ z0,
                          float* __restrict__ z, float* __restrict__ ldj, int n) {
  int i = blockIdx.x * blockDim.x + threadIdx.x;
  if (i < n) {
    float v = mean[i] + eps[i] * expf(0.5f * logvar[i]);
    z0[i] = v;
    z[i]  = v;
  }
  if (i < BATCH) ldj[i] = 0.f;
}

// One Sylvester-flow step k (launched 4x sequentially).  One block per batch b.
__global__ __launch_bounds__(256) void flow_step_k(
    const float* __restrict__ amat,   // 1024 x (64*32), post-orthogonalization
    const float* __restrict__ fulld,  // B x (D*D*K)
    const float* __restrict__ d1t,    // B x (D*K), tanh applied
    const float* __restrict__ d2t,    // B x (D*K), tanh applied
    const float* __restrict__ bfl,    // B x (D*K)
    float* __restrict__ zvec,         // B x 64 (updated in place)
    float* __restrict__ ldj,          // B
    int k) {
  int b = blockIdx.x, t = threadIdx.x;
  __shared__ float q[ZDIM * DDIM];
  __shared__ float r1[DDIM * DDIM];
  __shared__ float r2[DDIM * DDIM];
  __shared__ float qr1[ZDIM * DDIM];
  __shared__ float qr2[ZDIM * DDIM];
  __shared__ float zl[ZDIM];
  __shared__ float th[DDIM];
  __shared__ float dj[DDIM];

  const float* qg = amat + (size_t)(b * KF + k) * (ZDIM * DDIM);
  // async preload of q tile (pure copy -> async mover path)
  for (int i = t; i < ZDIM * DDIM; i += 256) async_g2l_b32(qg + i, &q[i]);
  for (int i = t; i < DDIM * DDIM; i += 256) {
    int d = i >> 5, e = i & 31;
    float fd_de = fulld[(size_t)b * (DDIM * DDIM * KF) + (size_t)i * KF + k];
    float fd_ed = fulld[(size_t)b * (DDIM * DDIM * KF) +
                        (size_t)(e * DDIM + d) * KF + k];
    float dg1 = d1t[(b * DDIM + d) * KF + k];
    float dg2 = d2t[(b * DDIM + d) * KF + k];
    r1[i] = (d < e) ? fd_de : ((d == e) ? dg1 : 0.f);
    r2[i] = (d < e) ? fd_ed : ((d == e) ? dg2 : 0.f);
  }
  if (t < ZDIM) zl[t] = zvec[b * ZDIM + t];
  async_wait0();
  __syncthreads();

  // qr2[z][e] = sum_d q[z][d]*r2[e][d];  qr1[z][e] = sum_d q[z][d]*r1[d][e]
  for (int i = t; i < ZDIM * DDIM; i += 256) {
    int z = i >> 5, e = i & 31;
    float a1 = 0.f, a2 = 0.f;
    for (int d = 0; d < DDIM; d++) {
      float qv = q[z * DDIM + d];
      a2 += qv * r2[e * DDIM + d];
      a1 += qv * r1[d * DDIM + e];
    }
    qr2[i] = a2;
    qr1[i] = a1;
  }
  __syncthreads();

  if (t < DDIM) {
    float a = 0.f;
    for (int z = 0; z < ZDIM; z++) a += zl[z] * qr2[z * DDIM + t];
    a += bfl[(b * DDIM + t) * KF + k];
    float tv = tanhf(a);
    th[t] = tv;
    float dr1 = d1t[(b * DDIM + t) * KF + k];
    float dr2 = d2t[(b * DDIM + t) * KF + k];
    float djv = (1.f - tv * tv) * (dr1 * dr2) + 1.f;
    dj[t] = logf(fabsf(djv));
  }
  __syncthreads();

  if (t < ZDIM) {
    float a = 0.f;
    for (int e = 0; e < DDIM; e++) a += th[e] * qr1[t * DDIM + e];
    zvec[b * ZDIM + t] = zl[t] + a;
  }
  if (t == 0) {
    float s = 0.f;
    for (int e = 0; e < DDIM; e++) s += dj[e];
    ldj[b] += s;
  }
}

// ---------------------------------------------------------------------------
// Host launch
// ---------------------------------------------------------------------------
extern "C" void kernel_launch(void* const* d_in, const int* in_sizes, int n_in,
                              void* d_out, int out_size, void* d_ws,
                              size_t ws_size, hipStream_t stream) {
  (void)in_sizes; (void)n_in; (void)out_size; (void)ws_size;
  const float* x     = (const float*)d_in[0];
  const float* eps   = (const float*)d_in[1];
  const float* cw1   = (const float*)d_in[2];
  const float* cb1   = (const float*)d_in[3];
  const float* g1    = (const float*)d_in[4];
  const float* be1   = (const float*)d_in[5];
  const float* cw2   = (const float*)d_in[6];
  const float* cb2   = (const float*)d_in[7];
  const float* g2    = (const float*)d_in[8];
  const float* be2   = (const float*)d_in[9];
  const float* cw3   = (const float*)d_in[10];
  const float* cb3   = (const float*)d_in[11];
  const float* g3    = (const float*)d_in[12];
  const float* be3   = (const float*)d_in[13];
  const float* d1_w  = (const float*)d_in[14];
  const float* d1_b  = (const float*)d_in[15];
  const float* bn1_g = (const float*)d_in[16];
  const float* bn1_b = (const float*)d_in[17];
  const float* mu_w  = (const float*)d_in[18];
  const float* mu_b  = (const float*)d_in[19];
  const float* lv_w  = (const float*)d_in[20];
  const float* lv_b  = (const float*)d_in[21];
  const float* ad_w  = (const float*)d_in[22];
  const float* ad_b  = (const float*)d_in[23];
  const float* adg1_w = (const float*)d_in[24];
  const float* adg1_b = (const float*)d_in[25];
  const float* adg2_w = (const float*)d_in[26];
  const float* adg2_b = (const float*)d_in[27];
  const float* aq_w  = (const float*)d_in[28];
  const float* aq_b  = (const float*)d_in[29];
  const float* ab_w  = (const float*)d_in[30];
  const float* ab_b  = (const float*)d_in[31];
  const float* d3_w  = (const float*)d_in[32];
  const float* d3_b  = (const float*)d_in[33];
  const float* bn3_g = (const float*)d_in[34];
  const float* bn3_b = (const float*)d_in[35];
  const float* d4_w  = (const float*)d_in[36];
  const float* d4_b  = (const float*)d_in[37];
  const float* bn4_g = (const float*)d_in[38];
  const float* bn4_b = (const float*)d_in[39];
  const float* tw1   = (const float*)d_in[40];
  const float* tb1   = (const float*)d_in[41];
  const float* tg1   = (const float*)d_in[42];
  const float* tbe1  = (const float*)d_in[43];
  const float* tw2   = (const float*)d_in[44];
  const float* tb2   = (const float*)d_in[45];
  const float* tg2   = (const float*)d_in[46];
  const float* tbe2  = (const float*)d_in[47];
  const float* tw3   = (const float*)d_in[48];
  const float* tb3   = (const float*)d_in[49];

  float* out = (float*)d_out;
  // output layout: x_dec | mean | logvar | ldj | z0 | z
  float* o_xdec = out;                 // 256*26*4 = 26624
  float* o_mean = out + 26624;         // 16384
  float* o_lv   = out + 43008;         // 16384
  float* o_ldj  = out + 59392;         // 256
  float* o_z0   = out + 59648;         // 16384
  float* o_z    = out + 76032;         // 16384

  float* W = (float*)d_ws;
  float* c1    = W;               // 256*32*24
  float* c2    = c1 + 196608;     // 256*16*20
  float* c3    = c2 + 81920;      // 256*8*14  (== flat 256x112)
  float* d1o   = c3 + 28672;      // 256*1024
  float* fulld = d1o + 262144;    // 256*4096
  float* dg1   = fulld + 1048576; // 256*128
  float* dg2   = dg1 + 32768;
  float* bfl   = dg2 + 32768;
  float* qb    = bfl + 32768;     // 256*8192 (== amat 1024 x 2048)
  float* d3o   = qb + 2097152;    // 256*1024
  float* d4o   = d3o + 262144;    // 256*112
  float* t1o   = d4o + 28672;     // 256*16*20
  float* t2o   = t1o + 81920;     // 256*32*24

  // ---- encoder ----
  conv2d_k<<<(196608 + 255) / 256, 256, 0, stream>>>(x, cw1, cb1, c1, BATCH, 1, 26, 4, 32, 3, 4, 24);
  bn2d_relu_k<<<32, 256, 0, stream>>>(c1, g1, be1, c1, BATCH, 32, 24);
  conv2d_k<<<(81920 + 255) / 256, 256, 0, stream>>>(c1, cw2, cb2, c2, BATCH, 32, 24, 1, 16, 5, 1, 20);
  bn2d_relu_k<<<16, 256, 0, stream>>>(c2, g2, be2, c2, BATCH, 16, 20);
  conv2d_k<<<(28672 + 255) / 256, 256, 0, stream>>>(c2, cw3, cb3, c3, BATCH, 16, 20, 1, 8, 7, 1, 14);
  bn2d_relu_k<<<8, 256, 0, stream>>>(c3, g3, be3, c3, BATCH, 8, 14);

  // ---- d1 + bn1 ----
  gemm_wmma_f32<<<(BATCH / 16) * (QOUTD / 16), 32, 0, stream>>>(c3, d1_w, d1_b, d1o, BATCH, QOUTD, 112, 0);
  bn1d_relu_k<<<4, 256, 0, stream>>>(d1o, bn1_g, bn1_b, d1o, BATCH, QOUTD);

  // ---- heads ----
  gemm_wmma_f32<<<(BATCH / 16) * (LATD / 16), 32, 0, stream>>>(d1o, mu_w, mu_b, o_mean, BATCH, LATD, QOUTD, 0);
  gemm_wmma_f32<<<(BATCH / 16) * (LATD / 16), 32, 0, stream>>>(d1o, lv_w, lv_b, o_lv, BATCH, LATD, QOUTD, 0);
  gemm_wmma_f32<<<(BATCH / 16) * (4096 / 16), 32, 0, stream>>>(d1o, ad_w, ad_b, fulld, BATCH, 4096, QOUTD, 0);
  gemm_wmma_f32<<<(BATCH / 16) * (128 / 16), 32, 0, stream>>>(d1o, adg1_w, adg1_b, dg1, BATCH, 128, QOUTD, 1);
  gemm_wmma_f32<<<(BATCH / 16) * (128 / 16), 32, 0, stream>>>(d1o, adg2_w, adg2_b, dg2, BATCH, 128, QOUTD, 1);
  gemm_wmma_f32<<<(BATCH / 16) * (8192 / 16), 32, 0, stream>>>(d1o, aq_w, aq_b, qb, BATCH, 8192, QOUTD, 0);
  gemm_wmma_f32<<<(BATCH / 16) * (128 / 16), 32, 0, stream>>>(d1o, ab_w, ab_b, bfl, BATCH, 128, QOUTD, 0);

  // ---- normalize rows of q, then 100-step orthogonalization (WMMA f32) ----
  row_norm_k<<<NMAT, 256, 0, stream>>>(qb);
  ortho_scan<<<NMAT / 2, 64, 0, stream>>>(qb);

  // ---- reparameterize + flows ----
  z0_init_k<<<(BATCH * LATD + 255) / 256, 256, 0, stream>>>(o_mean, o_lv, eps, o_z0, o_z, o_ldj, BATCH * LATD);
  for (int k = 0; k < KF; k++)
    flow_step_k<<<BATCH, 256, 0, stream>>>(qb, fulld, dg1, dg2, bfl, o_z, o_ldj, k);

  // ---- decoder ----
  gemm_wmma_f32<<<(BATCH / 16) * (QOUTD / 16), 32, 0, stream>>>(o_z, d3_w, d3_b, d3o, BATCH, QOUTD, LATD, 0);
  bn1d_relu_k<<<4, 256, 0, stream>>>(d3o, bn3_g, bn3_b, d3o, BATCH, QOUTD);
  gemm_wmma_f32<<<(BATCH / 16) * (112 / 16), 32, 0, stream>>>(d3o, d4_w, d4_b, d4o, BATCH, 112, QOUTD, 0);
  bn1d_relu_k<<<1, 256, 0, stream>>>(d4o, bn4_g, bn4_b, d4o, BATCH, 112);

  tconv2d_k<<<(81920 + 255) / 256, 256, 0, stream>>>(d4o, tw1, tb1, t1o, BATCH, 8, 14, 1, 16, 7, 1, 20, 1);
  bn2d_relu_k<<<16, 256, 0, stream>>>(t1o, tg1, tbe1, t1o, BATCH, 16, 20);
  tconv2d_k<<<(196608 + 255) / 256, 256, 0, stream>>>(t1o, tw2, tb2, t2o, BATCH, 16, 20, 1, 32, 5, 1, 24, 1);
  bn2d_relu_k<<<32, 256, 0, stream>>>(t2o, tg2, tbe2, t2o, BATCH, 32, 24);
  tconv2d_k<<<(26624 + 255) / 256, 256, 0, stream>>>(t2o, tw3, tb3, o_xdec, BATCH, 32, 24, 1, 1, 3, 4, 26, 4);
}